// LengthRegulator_21174188769595
// MI455X (gfx1250) — compile-verified
//
#include <hip/hip_runtime.h>
#include <hip/hip_bf16.h>
#include <stdint.h>

// ---------------------------------------------------------------------------
// LengthRegulator, MI455X (gfx1250).
//
// The reference's duration-predictor (conv/LN/linear stack) is DEAD CODE:
// `durations = target_durations` overrides the prediction and the returned
// tuple only depends on (x, target_durations). The live work is a cumsum and
// a 256 MB row-gather/zero-fill -> purely HBM-bandwidth-bound:
//   stores: 8*8192*1024*4B = 256 MB   (each output row written exactly once)
//   loads : x = 32 MB (fits in 192 MB L2; gather reuse hits L2)
//   => ~288 MB / 23.3 TB/s  ~ 12.4 us roofline.
// Output stream is never re-read -> non-temporal stores keep x/cum in cache.
// CDNA5 path exercised: GLOBAL_LOAD_ASYNC_TO_LDS_B128 (ASYNCcnt) to stage the
// per-batch cumsum row into LDS, drained with s_wait_asynccnt.
// ---------------------------------------------------------------------------

#define B_    8
#define T_    1024
#define D_    1024
#define MAXF  8192            // T * DMAX

typedef float v4f __attribute__((ext_vector_type(4)));
typedef int   v4i __attribute__((ext_vector_type(4)));

// Address-space-qualified views for the async-to-LDS builtin, whose exact
// signature (from the round-1 diagnostic) is:
//   void (v4i addrspace(1)* gsrc, v4i addrspace(3)* ldsdst, imm off, imm cpol)
typedef __attribute__((address_space(1))) v4i gv4i;   // global
typedef __attribute__((address_space(3))) v4i sv4i;   // LDS

#ifndef __has_builtin
#define __has_builtin(x) 0
#endif

#if __has_builtin(__builtin_amdgcn_global_load_async_to_lds_b128)
#define HAVE_ASYNC_LDS 1
#else
#define HAVE_ASYNC_LDS 0
#endif

__device__ __forceinline__ void wait_asynccnt0() {
#if __has_builtin(__builtin_amdgcn_s_wait_asynccnt)
  __builtin_amdgcn_s_wait_asynccnt(0);
#else
  asm volatile("s_wait_asynccnt 0x0" ::: "memory");
#endif
}

// ---------------------------------------------------------------------------
// Kernel 1: per-batch inclusive cumsum of target_durations (int32), plus
// pass-through of durations (as float) into the tail of d_out.
// One block per batch row; 256 threads x 4 elements each.
// ---------------------------------------------------------------------------
__global__ void __launch_bounds__(256)
lr_cumsum_kernel(const int* __restrict__ dur, int* __restrict__ cum,
                 float* __restrict__ out_dur) {
  __shared__ int part[256];
  const int b = blockIdx.x;
  const int t = threadIdx.x;
  const int base = b * T_ + t * 4;

  v4i d = *(const v4i*)(dur + base);
  const int s0 = d.x;
  const int s1 = s0 + d.y;
  const int s2 = s1 + d.z;
  const int s3 = s2 + d.w;

  part[t] = s3;
  __syncthreads();
  // Hillis-Steele inclusive scan over the 256 per-thread sums.
  #pragma unroll
  for (int off = 1; off < 256; off <<= 1) {
    const int v   = part[t];
    const int add = (t >= off) ? part[t - off] : 0;
    __syncthreads();
    part[t] = v + add;
    __syncthreads();
  }
  const int pref = (t == 0) ? 0 : part[t - 1];

  v4i c; c.x = pref + s0; c.y = pref + s1; c.z = pref + s2; c.w = pref + s3;
  *(v4i*)(cum + base) = c;

  // durations output (reference returns int32; harness exposes as float*)
  v4f fd; fd.x = (float)d.x; fd.y = (float)d.y; fd.z = (float)d.z; fd.w = (float)d.w;
  *(v4f*)(out_dur + base) = fd;
}

// ---------------------------------------------------------------------------
// Kernel 2: length regulation (gather + zero-fill).
// Grid: (MAXF/32, B). Block: 256 threads = 8 waves; each wave handles 4
// consecutive output frames. cum[b][0..1023] staged in LDS (async-to-LDS on
// CDNA5), then each wave does a uniform binary search (searchsorted "right")
// and streams the 4 KB row with float4 non-temporal stores.
// ---------------------------------------------------------------------------
__global__ void __launch_bounds__(256)
lr_expand_kernel(const float* __restrict__ x, const int* __restrict__ cum,
                 float* __restrict__ out) {
  __shared__ int shcum[T_] __attribute__((aligned(16)));

  const int b    = blockIdx.y;
  const int t    = threadIdx.x;       // 0..255
  const int lane = t & 31;
  const int wave = t >> 5;

  const int* crow = cum + b * T_;
#if HAVE_ASYNC_LDS
  // 256 lanes x b128 = the whole 4 KB cum row staged via the async-DMA path
  // (no VGPR round trip), tracked by ASYNCcnt.
  __builtin_amdgcn_global_load_async_to_lds_b128(
      (gv4i*)(crow + t * 4),
      (sv4i*)(&shcum[t * 4]),
      0, 0);
  wait_asynccnt0();
#else
  *(v4i*)(&shcum[t * 4]) = *(const v4i*)(crow + t * 4);
#endif
  __syncthreads();

  const int total = shcum[T_ - 1];
  const int fbase = blockIdx.x * 32 + wave * 4;

  #pragma unroll
  for (int i = 0; i < 4; ++i) {
    const int f = fbase + i;

    // searchsorted(cum, f, side="right"): first idx with cum[idx] > f.
    // Uniform across the wave -> LDS broadcast reads, no divergence.
    int lo = 0, hi = T_;
    while (lo < hi) {
      const int mid = (lo + hi) >> 1;
      if (shcum[mid] <= f) lo = mid + 1; else hi = mid;
    }
    const int tok = (lo < T_) ? lo : (T_ - 1);

    v4f* __restrict__ dst =
        (v4f*)(out + ((size_t)b * MAXF + (size_t)f) * D_);

    if (f < total) {
      const v4f* __restrict__ src =
          (const v4f*)(x + ((size_t)b * T_ + (size_t)tok) * D_);
      #pragma unroll
      for (int j = 0; j < 8; ++j) {
        const v4f v = src[j * 32 + lane];           // cached (x reused ~3.5x)
        __builtin_nontemporal_store(v, dst + j * 32 + lane);  // streaming out
      }
    } else {
      const v4f z = {0.f, 0.f, 0.f, 0.f};
      #pragma unroll
      for (int j = 0; j < 8; ++j)
        __builtin_nontemporal_store(z, dst + j * 32 + lane);
    }
  }
}

// ---------------------------------------------------------------------------
extern "C" void kernel_launch(void* const* d_in, const int* in_sizes, int n_in,
                              void* d_out, int out_size, void* d_ws, size_t ws_size,
                              hipStream_t stream) {
  (void)in_sizes; (void)n_in; (void)out_size; (void)ws_size;

  // setup_inputs() order: x, target_durations, w1, b1, g1, be1, w2, b2, g2,
  // be2, wl, bl. The predictor weights (indices 2..11) are dead code.
  const float* x   = (const float*)d_in[0];
  const int*   dur = (const int*)  d_in[1];

  float* out     = (float*)d_out;                       // [B, MAXF, D]
  float* out_dur = out + (size_t)B_ * MAXF * D_;        // [B, T]
  int*   cum     = (int*)d_ws;                          // [B, T] scratch

  lr_cumsum_kernel<<<B_, 256, 0, stream>>>(dur, cum, out_dur);
  lr_expand_kernel<<<dim3(MAXF / 32, B_), 256, 0, stream>>>(x, cum, out);
}